// RoipoolingLayer_90142773609315
// MI455X (gfx1250) — compile-verified
//
#include <hip/hip_runtime.h>
#include <math.h>

// RoI pool (crop_and_resize 14x14 bilinear + 2x2 max -> 7x7), f32.
// feat: [64,64,512], rois: [N,4] (y1,x1,y2,x2 normalized), out: [N,7,7,512].

#define FH    64
#define FW    64
#define FC    512
#define CROPN 14
#define POOLN 7
#define TPB   128   // 128 threads * float4 = 512 channels

// CDNA5 async global->LDS path (ASYNCcnt). Guarded so host pass / older
// toolchains fall back to direct global loads.
#if defined(__HIP_DEVICE_COMPILE__) && __has_builtin(__builtin_amdgcn_global_load_async_to_lds_b128)
#define HAS_ASYNC_LDS 1
#else
#define HAS_ASYNC_LDS 0
#endif

#if defined(__HIP_DEVICE_COMPILE__) && __has_builtin(__builtin_amdgcn_s_wait_asynccnt)
#define WAIT_ASYNCCNT(n) __builtin_amdgcn_s_wait_asynccnt(n)
#else
#define WAIT_ASYNCCNT(n) asm volatile("s_wait_asynccnt %0" ::"i"(n) : "memory")
#endif

#if HAS_ASYNC_LDS
// Builtin signature (from clang diagnostic): param0 is a non-const pointer to
// int __attribute__((vector_size(16))) in addrspace(1); param1 in addrspace(3).
typedef int v4i_t __attribute__((vector_size(16)));
typedef __attribute__((address_space(1))) v4i_t as1_v4i;
typedef __attribute__((address_space(3))) v4i_t as3_v4i;

__device__ __forceinline__ void async_copy_b128(const float* g, float* l) {
  __builtin_amdgcn_global_load_async_to_lds_b128((as1_v4i*)g, (as3_v4i*)l, 0, 0);
}
#endif

__device__ __forceinline__ float4 bilerp4(float4 a, float4 b, float4 c, float4 d,
                                          float wx, float wy) {
  const float owx = 1.0f - wx, owy = 1.0f - wy;
  float4 r;
  r.x = (a.x * owx + b.x * wx) * owy + (c.x * owx + d.x * wx) * wy;
  r.y = (a.y * owx + b.y * wx) * owy + (c.y * owx + d.y * wx) * wy;
  r.z = (a.z * owx + b.z * wx) * owy + (c.z * owx + d.z * wx) * wy;
  r.w = (a.w * owx + b.w * wx) * owy + (c.w * owx + d.w * wx) * wy;
  return r;
}

__device__ __forceinline__ float4 vmax4(float4 a, float4 b) {
  float4 r;
  r.x = fmaxf(a.x, b.x);
  r.y = fmaxf(a.y, b.y);
  r.z = fmaxf(a.z, b.z);
  r.w = fmaxf(a.w, b.w);
  return r;
}

__global__ __launch_bounds__(TPB) void roi_pool_kernel(
    const float* __restrict__ feat, const float* __restrict__ rois,
    float* __restrict__ out, int N) {
  const int bx  = blockIdx.x;
  const int n   = bx / POOLN;
  const int py  = bx - n * POOLN;
  const int tid = (int)threadIdx.x;
  const int c4  = tid * 4;

  const float ry1 = rois[n * 4 + 0];
  const float rx1 = rois[n * 4 + 1];
  const float ry2 = rois[n * 4 + 2];
  const float rx2 = rois[n * 4 + 3];

  // y-corner data for the two crop rows (iy = 2*py, 2*py+1) of this block.
  int   y0i[2], y1i[2];
  float wyv[2];
#pragma unroll
  for (int r = 0; r < 2; ++r) {
    const float t  = (float)(2 * py + r) * (1.0f / (CROPN - 1));
    const float ys = (ry1 + t * (ry2 - ry1)) * (float)(FH - 1);
    const float yf = fminf(fmaxf(floorf(ys), 0.0f), (float)(FH - 1));
    y0i[r] = (int)yf;
    y1i[r] = min(y0i[r] + 1, FH - 1);
    wyv[r] = ys - yf;  // matches reference: frac against clipped floor
  }

  auto xcorner = [&](int ix, int& x0, int& x1, float& wx) {
    const float t  = (float)ix * (1.0f / (CROPN - 1));
    const float xs = (rx1 + t * (rx2 - rx1)) * (float)(FW - 1);
    const float xf = fminf(fmaxf(floorf(xs), 0.0f), (float)(FW - 1));
    x0 = (int)xf;
    x1 = min(x0 + 1, FW - 1);
    wx = xs - xf;
  };

  float* outp = out + ((size_t)n * POOLN + (size_t)py) * POOLN * FC + c4;

#if HAS_ASYNC_LDS
  // Double-buffered async staging: 16 corner row-segments (2KB each) per
  // pooled pixel. Each lane copies and later reads back its own 16 bytes per
  // segment, so wave-local ASYNCcnt is the only sync needed (no barrier).
  __shared__ __align__(16) float lds[2][16][FC];

  auto stage = [&](int px, int buf) {
#pragma unroll
    for (int j = 0; j < 2; ++j) {
      int x0, x1;
      float wx;
      xcorner(2 * px + j, x0, x1, wx);
      (void)wx;
#pragma unroll
      for (int r = 0; r < 2; ++r) {
        const int    s   = (r * 2 + j) * 4;
        const float* g00 = feat + (size_t)(y0i[r] * FW + x0) * FC + c4;
        const float* g01 = feat + (size_t)(y0i[r] * FW + x1) * FC + c4;
        const float* g10 = feat + (size_t)(y1i[r] * FW + x0) * FC + c4;
        const float* g11 = feat + (size_t)(y1i[r] * FW + x1) * FC + c4;
        async_copy_b128(g00, &lds[buf][s + 0][c4]);
        async_copy_b128(g01, &lds[buf][s + 1][c4]);
        async_copy_b128(g10, &lds[buf][s + 2][c4]);
        async_copy_b128(g11, &lds[buf][s + 3][c4]);
      }
    }
  };

  stage(0, 0);
  for (int px = 0; px < POOLN; ++px) {
    const int buf = px & 1;
    if (px + 1 < POOLN) {
      stage(px + 1, buf ^ 1);   // prefetch next pixel (16 more in flight)
      WAIT_ASYNCCNT(16);        // in-order: oldest 16 (current pixel) done
    } else {
      WAIT_ASYNCCNT(0);
    }

    float4 acc = make_float4(-__builtin_inff(), -__builtin_inff(),
                             -__builtin_inff(), -__builtin_inff());
#pragma unroll
    for (int j = 0; j < 2; ++j) {
      int x0, x1;
      float wx;
      xcorner(2 * px + j, x0, x1, wx);
      (void)x0;
      (void)x1;
#pragma unroll
      for (int r = 0; r < 2; ++r) {
        const int    s   = (r * 2 + j) * 4;
        const float4 f00 = *(const float4*)&lds[buf][s + 0][c4];
        const float4 f01 = *(const float4*)&lds[buf][s + 1][c4];
        const float4 f10 = *(const float4*)&lds[buf][s + 2][c4];
        const float4 f11 = *(const float4*)&lds[buf][s + 3][c4];
        acc = vmax4(acc, bilerp4(f00, f01, f10, f11, wx, wyv[r]));
      }
    }
    *(float4*)(outp + (size_t)px * FC) = acc;
  }
#else
  // Fallback: direct coalesced global float4 gathers (L2-resident feature map).
  for (int px = 0; px < POOLN; ++px) {
    float4 acc = make_float4(-__builtin_inff(), -__builtin_inff(),
                             -__builtin_inff(), -__builtin_inff());
#pragma unroll
    for (int j = 0; j < 2; ++j) {
      int x0, x1;
      float wx;
      xcorner(2 * px + j, x0, x1, wx);
#pragma unroll
      for (int r = 0; r < 2; ++r) {
        const float4 f00 = *(const float4*)(feat + (size_t)(y0i[r] * FW + x0) * FC + c4);
        const float4 f01 = *(const float4*)(feat + (size_t)(y0i[r] * FW + x1) * FC + c4);
        const float4 f10 = *(const float4*)(feat + (size_t)(y1i[r] * FW + x0) * FC + c4);
        const float4 f11 = *(const float4*)(feat + (size_t)(y1i[r] * FW + x1) * FC + c4);
        acc = vmax4(acc, bilerp4(f00, f01, f10, f11, wx, wyv[r]));
      }
    }
    *(float4*)(outp + (size_t)px * FC) = acc;
  }
#endif
}

extern "C" void kernel_launch(void* const* d_in, const int* in_sizes, int n_in,
                              void* d_out, int out_size, void* d_ws, size_t ws_size,
                              hipStream_t stream) {
  (void)n_in;
  (void)out_size;
  (void)d_ws;
  (void)ws_size;
  const float* feat = (const float*)d_in[0];  // [1,64,64,512] f32
  const float* rois = (const float*)d_in[1];  // [N,4] f32
  float*       out  = (float*)d_out;          // [N,7,7,512] f32
  const int    N    = in_sizes[1] / 4;

  dim3 grid(N * POOLN);
  dim3 block(TPB);
  roi_pool_kernel<<<grid, block, 0, stream>>>(feat, rois, out, N);
}